// graphSAGELayer_46772193853697
// MI455X (gfx1250) — compile-verified
//
#include <hip/hip_runtime.h>
#include <hip/hip_bf16.h>

typedef __attribute__((ext_vector_type(2))) float v2f;
typedef __attribute__((ext_vector_type(8))) float v8f;

#define DIM   256   // feature dim D
#define DOUT  256   // output dim
#define KNBR  16    // neighbors per node
#define TILE_M 16   // nodes per block
#define LDSP  260   // padded LDS row stride (floats) to avoid bank conflicts

__global__ __launch_bounds__(256) void sage_fused_kernel(
    const float* __restrict__ h,
    const int*   __restrict__ nbr,
    const float* __restrict__ W,
    float*       __restrict__ out,
    int nnodes)
{
    __shared__ float aggLDS[TILE_M * LDSP];   // 16x256 agg tile (padded rows)
    __shared__ float zLDS[TILE_M * LDSP];     // 16x256 z tile (padded rows)
    __shared__ float normLDS[TILE_M];         // per-row sum of squares

    const int t    = threadIdx.x;
    const int r    = t >> 4;          // row within tile, 0..15
    const int cg   = t & 15;          // column group, 0..15 (16 floats each)
    const int node = blockIdx.x * TILE_M + r;
    const int nodeC = node < nnodes ? node : (nnodes - 1);  // clamp for loads

    // ---------------- Phase 1: gather + mean aggregate ----------------
    float acc[16];
    {
        const float* hrow = h + (size_t)nodeC * DIM + cg * 16;
        #pragma unroll
        for (int i = 0; i < 4; ++i) {
            float4 v = ((const float4*)hrow)[i];
            acc[4*i+0] = v.x; acc[4*i+1] = v.y; acc[4*i+2] = v.z; acc[4*i+3] = v.w;
        }
        const int* nrow = nbr + (size_t)nodeC * KNBR;
        #pragma unroll 4
        for (int k = 0; k < KNBR; ++k) {
            const int src = nrow[k];
            const float* srow = h + (size_t)src * DIM + cg * 16;
            #pragma unroll
            for (int i = 0; i < 4; ++i) {
                float4 v = ((const float4*)srow)[i];
                acc[4*i+0] += v.x; acc[4*i+1] += v.y;
                acc[4*i+2] += v.z; acc[4*i+3] += v.w;
            }
        }
        const float inv = 1.0f / (float)(KNBR + 1);
        #pragma unroll
        for (int i = 0; i < 16; ++i)
            aggLDS[r * LDSP + cg * 16 + i] = acc[i] * inv;
    }
    __syncthreads();

    // ---------------- Phase 2: GEMM via V_WMMA_F32_16X16X4_F32 ----------------
    // wave w computes output col tiles [32w,32w+16) and [32w+16,32w+32)
    const int lane = t & 31;
    const int wave = t >> 5;
    const int m    = lane & 15;       // row (A) / col-in-tile (B,C)
    const int hi   = lane >> 4;       // half-wave select

    v8f c0 = {};
    v8f c1 = {};
    const float* wbase = W + (size_t)(wave * 32 + m);  // column offset into W rows

    #pragma unroll 4
    for (int k0 = 0; k0 < DIM; k0 += 4) {
        const int kk = k0 + 2 * hi;
        // A frag: lane(m,hi) holds agg[m][kk], agg[m][kk+1]  (ISA 16x4 f32 layout)
        const float* ap = &aggLDS[m * LDSP + kk];
        v2f a; a.x = ap[0]; a.y = ap[1];
        // B frags: lane(m,hi) holds W[kk][n], W[kk+1][n] for n = 32w+m (+16 for tile1)
        const float* wp = wbase + (size_t)kk * DOUT;
        v2f b0; b0.x = wp[0];  b0.y = wp[DOUT];
        v2f b1; b1.x = wp[16]; b1.y = wp[DOUT + 16];
        c0 = __builtin_amdgcn_wmma_f32_16x16x4_f32(false, a, false, b0, (short)0, c0, false, false);
        c1 = __builtin_amdgcn_wmma_f32_16x16x4_f32(false, a, false, b1, (short)0, c1, false, false);
    }

    // ---------------- Phase 3: LeakyReLU + stage z to LDS ----------------
    // C/D layout: VGPR v, lanes 0-15 -> row v, lanes 16-31 -> row v+8; col = lane%16
    #pragma unroll
    for (int v = 0; v < 8; ++v) {
        const int row = v + 8 * hi;
        float z0 = c0[v]; z0 = (z0 >= 0.0f) ? z0 : 0.2f * z0;
        float z1 = c1[v]; z1 = (z1 >= 0.0f) ? z1 : 0.2f * z1;
        zLDS[row * LDSP + wave * 32 + m]      = z0;
        zLDS[row * LDSP + wave * 32 + 16 + m] = z1;
    }
    if (t < TILE_M) normLDS[t] = 0.0f;
    __syncthreads();

    // ---------------- Phase 4: per-row L2 norm ----------------
    float ssq = 0.0f;
    #pragma unroll
    for (int i = 0; i < 16; ++i) {
        const float z = zLDS[r * LDSP + cg * 16 + i];
        ssq += z * z;
    }
    atomicAdd(&normLDS[r], ssq);   // ds_add_f32
    __syncthreads();

    const float nrm   = sqrtf(normLDS[r]);
    const float scale = 1.0f / fmaxf(nrm, 1e-12f);

    if (node < nnodes) {
        float* op = out + (size_t)node * DOUT + cg * 16;
        #pragma unroll
        for (int i = 0; i < 4; ++i) {
            float4 v;
            v.x = zLDS[r * LDSP + cg * 16 + 4*i + 0] * scale;
            v.y = zLDS[r * LDSP + cg * 16 + 4*i + 1] * scale;
            v.z = zLDS[r * LDSP + cg * 16 + 4*i + 2] * scale;
            v.w = zLDS[r * LDSP + cg * 16 + 4*i + 3] * scale;
            ((float4*)op)[i] = v;
        }
    }
}

extern "C" void kernel_launch(void* const* d_in, const int* in_sizes, int n_in,
                              void* d_out, int out_size, void* d_ws, size_t ws_size,
                              hipStream_t stream) {
    const float* h   = (const float*)d_in[0];   // [N, 256] fp32
    const int*   nbr = (const int*)d_in[1];     // [N, 16]  int
    const float* W   = (const float*)d_in[2];   // [256, 256] fp32
    float*       out = (float*)d_out;           // [N, 256] fp32

    const int nnodes = in_sizes[0] / DIM;
    const int nblocks = (nnodes + TILE_M - 1) / TILE_M;
    sage_fused_kernel<<<nblocks, 256, 0, stream>>>(h, nbr, W, out, nnodes);
}